// CrossLevelAttention_88751204204558
// MI455X (gfx1250) — compile-verified
//
#include <hip/hip_runtime.h>
#include <math.h>

typedef __attribute__((ext_vector_type(2))) float v2f;
typedef __attribute__((ext_vector_type(8))) float v8f;

#define LNUM 4
#define NNODES 65536
#define DDIM 512
#define D4 (DDIM / 4)                 // 128 float4 per row
#define RBLK 128                      // partial-sum blocks per level
#define ROWS_PER_BLK (NNODES / RBLK)  // 512
#define LN_EPS 1e-5f
#define NEG_INF -1e30f
#define ATTN_SCALE 0.08838834764831845f   // (D/NUM_HEADS)^-0.5 = 128^-0.5

// ---------------------------------------------------------------------------
// Stage 1: per-level column partial sums of x (L,N,D). Deterministic 2-stage
// reduction (no float atomics). 512 MB streaming read -> bandwidth bound.
// Each block streams a contiguous 1 MB slab (512 rows x 2 KB).
// ---------------------------------------------------------------------------
__global__ void cla_colsum_part(const float* __restrict__ x,
                                float* __restrict__ partial) {
    const int b  = blockIdx.x;          // 0 .. L*RBLK-1
    const int l  = b >> 7;              // RBLK == 128
    const int sb = b & 127;
    const int t  = threadIdx.x;         // 0..127, one float4 column each
    const float4* xr = (const float4*)x;
    size_t base = (size_t)(l * NNODES + sb * ROWS_PER_BLK) * D4 + t;
    float ax = 0.f, ay = 0.f, az = 0.f, aw = 0.f;
#pragma unroll 8
    for (int r = 0; r < ROWS_PER_BLK; ++r) {
        float4 vv = xr[base + (size_t)r * D4];
        ax += vv.x; ay += vv.y; az += vv.z; aw += vv.w;
    }
    float4 o; o.x = ax; o.y = ay; o.z = az; o.w = aw;
    ((float4*)partial)[(size_t)b * D4 + t] = o;
}

__global__ void cla_colsum_final(const float* __restrict__ partial,
                                 float* __restrict__ summ) {
    const int l = blockIdx.x;
    const int t = threadIdx.x;          // 0..127
    const float4* p = (const float4*)partial;
    float ax = 0.f, ay = 0.f, az = 0.f, aw = 0.f;
    for (int j = 0; j < RBLK; ++j) {    // fixed order -> deterministic
        float4 vv = p[(size_t)(l * RBLK + j) * D4 + t];
        ax += vv.x; ay += vv.y; az += vv.z; aw += vv.w;
    }
    const float inv = 1.0f / (float)NNODES;
    float4 o; o.x = ax * inv; o.y = ay * inv; o.z = az * inv; o.w = aw * inv;
    ((float4*)summ)[l * D4 + t] = o;
}

// ---------------------------------------------------------------------------
// Per-level matvec: vout[l,j] = (relu?) bias[l,j] + sum_d vin[l,d]*W[l,d,j]
// grid = LNUM*4 blocks of 128 threads; coalesced 512B W reads per step.
// ---------------------------------------------------------------------------
__global__ void cla_matvec512(const float* __restrict__ vin,
                              const float* __restrict__ W,
                              const float* __restrict__ bias,
                              float* __restrict__ vout,
                              int do_relu) {
    const int b = blockIdx.x;
    const int l = b >> 2;
    const int chunk = b & 3;
    const int t = threadIdx.x;
    const int j = chunk * 128 + t;
    __shared__ float s_in[DDIM];
    for (int i = t; i < DDIM; i += 128) s_in[i] = vin[l * DDIM + i];
    __syncthreads();
    const float* Wl = W + (size_t)l * DDIM * DDIM + j;
    float acc = bias[l * DDIM + j];
#pragma unroll 8
    for (int d = 0; d < DDIM; ++d)
        acc = fmaf(s_in[d], Wl[(size_t)d * DDIM], acc);
    if (do_relu) acc = fmaxf(acc, 0.f);
    vout[l * DDIM + j] = acc;
}

// ---------------------------------------------------------------------------
// Attention core, one wave32. f32 WMMA (exact precision):
//   scores = q @ k^T   : M=4(pad16), N=4(pad16), K=512 -> 128 chained WMMAs
//   softmax w/ diagonal mask (scalar, 4 lanes)
//   ctx    = attn @ v  : M=4(pad16), K=4, N=512 -> 32 WMMAs
// A 16x4 f32 layout: VGPR0 = K0 (lanes0-15) / K2 (lanes16-31), VGPR1 = K1/K3.
// B 4x16  f32 layout: lanes0-15 rows K0,K1 ; lanes16-31 rows K2,K3.
// Pad rows are zeroed by a float mask (no exec-masked loads, keeps EXEC full).
// ---------------------------------------------------------------------------
__global__ void cla_attn_core(const float* __restrict__ q,
                              const float* __restrict__ kmat,
                              const float* __restrict__ v,
                              float* __restrict__ ctx) {
    __shared__ float s_q[LNUM * DDIM];
    __shared__ float s_k[LNUM * DDIM];
    __shared__ float s_v[LNUM * DDIM];
    __shared__ float s_sc[16 * 16];
    __shared__ float s_at[LNUM * LNUM];

    const int lane = threadIdx.x;       // 0..31
    for (int i = lane; i < LNUM * DDIM; i += 32) {
        s_q[i] = q[i]; s_k[i] = kmat[i]; s_v[i] = v[i];
    }
    __syncthreads();

    const int half = lane >> 4;                    // 0 or 1
    const int mn   = lane & 15;                    // M index for A, N index for B
    const int rsafe = (mn < LNUM) ? mn : 0;        // clamped row (always in-bounds)
    const float rowmask = (mn < LNUM) ? 1.f : 0.f; // zero out pad rows via VALU mul
    const int koff  = half ? 2 : 0;

    // ---- scores = q @ k^T (accumulate over K=512 in steps of 4) ----
    v8f acc = {};
    const float* qrow = &s_q[rsafe * DDIM];
    const float* krow = &s_k[rsafe * DDIM];
    for (int kc = 0; kc < DDIM; kc += 4) {
        const int k0 = kc + koff;
        v2f a, b;
        a.x = qrow[k0]     * rowmask;   // unconditional ds_load, then mul
        a.y = qrow[k0 + 1] * rowmask;
        b.x = krow[k0]     * rowmask;   // B[k][n] = K[n][k]
        b.y = krow[k0 + 1] * rowmask;
        acc = __builtin_amdgcn_wmma_f32_16x16x4_f32(
            false, a, false, b, (short)0, acc, false, false);
    }
    // D layout: VGPR r holds row M=r+half*8, col = mn
#pragma unroll
    for (int r = 0; r < 8; ++r) s_sc[(r + half * 8) * 16 + mn] = acc[r];
    __syncthreads();

    // ---- masked softmax over each row (4 lanes do 4 rows) ----
    if (lane < LNUM) {
        float row[LNUM];
        float m = -3.4e38f;
        for (int j = 0; j < LNUM; ++j) {
            float s = (j == lane) ? NEG_INF : s_sc[lane * 16 + j] * ATTN_SCALE;
            row[j] = s; m = fmaxf(m, s);
        }
        float sum = 0.f;
        for (int j = 0; j < LNUM; ++j) { row[j] = expf(row[j] - m); sum += row[j]; }
        const float inv = 1.f / sum;
        for (int j = 0; j < LNUM; ++j) s_at[lane * LNUM + j] = row[j] * inv;
    }
    __syncthreads();

    // ---- ctx = attn @ v (single K=4 step per 16-wide N tile) ----
    v2f a2;
    a2.x = s_at[rsafe * LNUM + koff]     * rowmask;
    a2.y = s_at[rsafe * LNUM + koff + 1] * rowmask;
    for (int nt = 0; nt < DDIM; nt += 16) {
        v2f b2;
        b2.x = s_v[koff * DDIM + nt + mn];        // row K=koff
        b2.y = s_v[(koff + 1) * DDIM + nt + mn];  // row K=koff+1
        v8f c = {};
        c = __builtin_amdgcn_wmma_f32_16x16x4_f32(
            false, a2, false, b2, (short)0, c, false, false);
        if (half == 0) {
#pragma unroll
            for (int r = 0; r < LNUM; ++r)
                ctx[r * DDIM + nt + mn] = c[r];   // row M=r, col nt+mn
        }
    }
}

// ---------------------------------------------------------------------------
// Fused residual + LayerNorm: ONE WAVE32 PER ROW (8 rows / 256-thread block).
// Pure shuffle reduction: no LDS, no barriers. 1 GB traffic -> roofline kernel.
// ---------------------------------------------------------------------------
__global__ void cla_ln_fuse(const float* __restrict__ x,
                            const float* __restrict__ upd,
                            const float* __restrict__ gamma,
                            const float* __restrict__ beta,
                            float* __restrict__ out) {
    const int wid  = threadIdx.x >> 5;            // wave 0..7 in block
    const int lane = threadIdx.x & 31;
    const size_t row = (size_t)blockIdx.x * 8 + wid;   // 0 .. L*N-1
    const int l = (int)(row >> 16);               // N = 65536
    const float4* xr = (const float4*)x + row * D4;
    const float4* ur = (const float4*)upd + (size_t)l * D4;

    float4 y[4];
    float s = 0.f, ss = 0.f;
#pragma unroll
    for (int c = 0; c < 4; ++c) {
        float4 xv = xr[c * 32 + lane];
        float4 uv = ur[c * 32 + lane];
        float4 t;
        t.x = xv.x + uv.x; t.y = xv.y + uv.y;
        t.z = xv.z + uv.z; t.w = xv.w + uv.w;
        y[c] = t;
        s  += t.x + t.y + t.z + t.w;
        ss += t.x * t.x + t.y * t.y + t.z * t.z + t.w * t.w;
    }
#pragma unroll
    for (int o = 16; o > 0; o >>= 1) {            // wave32 reduction
        s  += __shfl_xor(s,  o, 32);
        ss += __shfl_xor(ss, o, 32);
    }
    const float mu   = s * (1.f / DDIM);
    const float var  = ss * (1.f / DDIM) - mu * mu;
    const float rstd = rsqrtf(var + LN_EPS);

    float4* orow = (float4*)out + row * D4;
    const float4* g4 = (const float4*)gamma + (size_t)l * D4;
    const float4* b4 = (const float4*)beta  + (size_t)l * D4;
#pragma unroll
    for (int c = 0; c < 4; ++c) {
        float4 g = g4[c * 32 + lane];
        float4 b = b4[c * 32 + lane];
        float4 t = y[c], o;
        o.x = (t.x - mu) * rstd * g.x + b.x;
        o.y = (t.y - mu) * rstd * g.y + b.y;
        o.z = (t.z - mu) * rstd * g.z + b.z;
        o.w = (t.w - mu) * rstd * g.w + b.w;
        orow[c * 32 + lane] = o;
    }
}

// ---------------------------------------------------------------------------
extern "C" void kernel_launch(void* const* d_in, const int* in_sizes, int n_in,
                              void* d_out, int out_size, void* d_ws, size_t ws_size,
                              hipStream_t stream) {
    const float* x     = (const float*)d_in[0];
    const float* Wq    = (const float*)d_in[1];
    const float* bq    = (const float*)d_in[2];
    const float* Wk    = (const float*)d_in[3];
    const float* bk    = (const float*)d_in[4];
    const float* Wv    = (const float*)d_in[5];
    const float* bv    = (const float*)d_in[6];
    const float* W1    = (const float*)d_in[7];
    const float* b1    = (const float*)d_in[8];
    const float* W2    = (const float*)d_in[9];
    const float* b2    = (const float*)d_in[10];
    const float* gamma = (const float*)d_in[11];
    const float* beta  = (const float*)d_in[12];
    float* out = (float*)d_out;

    float* ws      = (float*)d_ws;
    float* summ    = ws;                 // L*D = 2048
    float* q       = ws + 2048;
    float* k       = ws + 4096;
    float* v       = ws + 6144;
    float* ctx     = ws + 8192;
    float* h       = ws + 10240;
    float* upd     = ws + 12288;
    float* partial = ws + 14336;         // L*RBLK*D = 262144 floats

    cla_colsum_part <<<LNUM * RBLK, 128, 0, stream>>>(x, partial);
    cla_colsum_final<<<LNUM,        128, 0, stream>>>(partial, summ);

    cla_matvec512<<<LNUM * 4, 128, 0, stream>>>(summ, Wq, bq, q, 0);
    cla_matvec512<<<LNUM * 4, 128, 0, stream>>>(summ, Wk, bk, k, 0);
    cla_matvec512<<<LNUM * 4, 128, 0, stream>>>(summ, Wv, bv, v, 0);

    cla_attn_core<<<1, 32, 0, stream>>>(q, k, v, ctx);

    cla_matvec512<<<LNUM * 4, 128, 0, stream>>>(ctx, W1, b1, h, 1);
    cla_matvec512<<<LNUM * 4, 128, 0, stream>>>(h,   W2, b2, upd, 0);

    cla_ln_fuse<<<(LNUM * NNODES) / 8, 256, 0, stream>>>(x, upd, gamma, beta, out);
}